// NNAllpassFilter_28226525070335
// MI455X (gfx1250) — compile-verified
//
#include <hip/hip_runtime.h>
#include <hip/hip_bf16.h>

typedef float v2f __attribute__((ext_vector_type(2)));
typedef float v8f __attribute__((ext_vector_type(8)));

#define NB      256      // batch rows
#define TLEN    65536    // samples per row
#define DLY     2003     // delay length (static in reference)
#define PTILES  126      // ceil(2003 / 16) phase tiles per row
#define TAIL    1440     // TLEN - 32*DLY : phases that have a 33rd step

// One wave = one (row, 16-phase tile).  Y(16x16) = A'(16x16) x X(16x16) + C
// via four chained v_wmma_f32_16x16x4_f32 (K = 16 split 4x4).
//   A'[i][j] = fb^(i-1-j) (j<i), -1 (j==i), 0 (j>i)   -> folds y = bo - x
//   C[i][p]  = fb^i * s0[p]                            -> carry-in state
//   carry:     s' = fb*y15 + (1+fb)*x15
// A' entries come from a block-shared LDS table indexed by e = i-1-j:
//   tab[e+16] = 0 (e<-1), -1 (e==-1), fb^e (0<=e<=14)
__global__ __launch_bounds__(256) void allpass_wmma_kernel(
    const float* __restrict__ x, const float* __restrict__ fbp,
    float* __restrict__ y)
{
    __shared__ float tab[32];                        // indices 0..30 used

    const float fb = fbp[0];

    // ---- Build padded fb-power table once per block (31 entries)
    if (threadIdx.x < 31) {
        const int e = (int)threadIdx.x - 16;         // e in [-16, 14]
        float v;
        if (e < -1) {
            v = 0.f;                                 // above diagonal
        } else if (e == -1) {
            v = -1.f;                                // diagonal (-I term)
        } else {                                     // fb^e by squaring bits
            const float f1 = fb, f2 = fb * fb, f4 = f2 * f2, f8 = f4 * f4;
            v = ((e & 1) ? f1 : 1.f) * ((e & 2) ? f2 : 1.f) *
                ((e & 4) ? f4 : 1.f) * ((e & 8) ? f8 : 1.f);
        }
        tab[threadIdx.x] = v;
    }
    __syncthreads();

    const int b    = blockIdx.y;
    const int wave = threadIdx.x >> 5;
    const int pt   = blockIdx.x * 8 + wave;          // phase-tile index
    if (pt >= PTILES) return;                        // wave-uniform exit

    const int l  = threadIdx.x & 31;
    const int c  = l & 15;                           // column / phase-in-tile
    const int hi = l >> 4;                           // lane half
    const int p0 = pt << 4;
    const int p  = p0 + c;                           // absolute phase
    const bool storeok = (p < DLY);

    // ---- Gather A' in WMMA A-layout: lane half selects local K {0,1}/{2,3}
    //      A_q.{x,y} = A'[M=c][j = 4q + 2*hi + {0,1}],  e = c-1-j
    const int i0 = c + 15 - 2 * hi;                  // tab index for q=0,.x
    v2f A0, A1, A2, A3;
    A0.x = tab[i0     ];  A0.y = tab[i0 - 1 ];
    A1.x = tab[i0 - 4 ];  A1.y = tab[i0 - 5 ];
    A2.x = tab[i0 - 8 ];  A2.y = tab[i0 - 9 ];
    A3.x = tab[i0 - 12];  A3.y = tab[i0 - 13];

    // ---- Per-lane base pointers; all row offsets become immediate IOFFSETs
    const size_t rowb = (size_t)b * TLEN + (size_t)p0 + (size_t)c;
    const float* xb = x + rowb + (size_t)(2 * hi) * DLY;   // B rows 4q+2hi+cmp
    float*       yb = y + rowb + (size_t)(8 * hi) * DLY;   // D rows v+8hi

    const float fb2 = fb * fb, fb4 = fb2 * fb2, fb8 = fb4 * fb4;
    float s0 = 0.f;                                  // delay line starts at 0

    #pragma unroll
    for (int kt = 0; kt < 2; ++kt) {                 // two full 16-step tiles
        const int ko = kt * 16 * DLY;                // immediate-folded

        // ---- Load X tile in B-layout (constant offsets => single-instr VMEM)
        v2f B0, B1, B2, B3;
        B0.x = __builtin_nontemporal_load(xb + ko + 0 * DLY);
        B0.y = __builtin_nontemporal_load(xb + ko + 1 * DLY);
        B1.x = __builtin_nontemporal_load(xb + ko + 4 * DLY);
        B1.y = __builtin_nontemporal_load(xb + ko + 5 * DLY);
        B2.x = __builtin_nontemporal_load(xb + ko + 8 * DLY);
        B2.y = __builtin_nontemporal_load(xb + ko + 9 * DLY);
        B3.x = __builtin_nontemporal_load(xb + ko + 12 * DLY);
        B3.y = __builtin_nontemporal_load(xb + ko + 13 * DLY);

        // ---- C[i][p] = fb^(v + 8*hi) * s0  (pwv is per-lane: hi lanes start at fb^8)
        v8f acc;
        {
            float pwv = hi ? fb8 : 1.f;
            acc[0] = s0 * pwv; pwv *= fb;
            acc[1] = s0 * pwv; pwv *= fb;
            acc[2] = s0 * pwv; pwv *= fb;
            acc[3] = s0 * pwv; pwv *= fb;
            acc[4] = s0 * pwv; pwv *= fb;
            acc[5] = s0 * pwv; pwv *= fb;
            acc[6] = s0 * pwv; pwv *= fb;
            acc[7] = s0 * pwv;
        }

        // ---- Y = A'·X + C  (EXEC all-ones here; compiler handles hazards)
        acc = __builtin_amdgcn_wmma_f32_16x16x4_f32(false, A3, false, B3,
                                                    (short)0, acc, false, false);
        acc = __builtin_amdgcn_wmma_f32_16x16x4_f32(false, A2, false, B2,
                                                    (short)0, acc, false, false);
        acc = __builtin_amdgcn_wmma_f32_16x16x4_f32(false, A1, false, B1,
                                                    (short)0, acc, false, false);
        acc = __builtin_amdgcn_wmma_f32_16x16x4_f32(false, A0, false, B0,
                                                    (short)0, acc, false, false);

        // ---- Store Y tile (mask padded phases >= 2003), streaming
        if (storeok) {
            __builtin_nontemporal_store(acc[0], yb + ko + 0 * DLY);
            __builtin_nontemporal_store(acc[1], yb + ko + 1 * DLY);
            __builtin_nontemporal_store(acc[2], yb + ko + 2 * DLY);
            __builtin_nontemporal_store(acc[3], yb + ko + 3 * DLY);
            __builtin_nontemporal_store(acc[4], yb + ko + 4 * DLY);
            __builtin_nontemporal_store(acc[5], yb + ko + 5 * DLY);
            __builtin_nontemporal_store(acc[6], yb + ko + 6 * DLY);
            __builtin_nontemporal_store(acc[7], yb + ko + 7 * DLY);
        }

        // ---- Carry: hi lanes hold row 15 (acc[7]) and x15 (B3.y)
        float tmp = fb * acc[7] + (1.f + fb) * B3.y;
        s0 = __shfl(tmp, 16 + c, 32);                // broadcast per column
    }

    // ---- Epilogue: single step k=32 exists only for phases < 1440
    if (hi == 0 && p < TAIL) {
        const size_t t = (size_t)b * TLEN + (size_t)p + (size_t)(32 * DLY);
        const float xv = __builtin_nontemporal_load(x + t);
        __builtin_nontemporal_store(s0 - xv, y + t);
    }
}

extern "C" void kernel_launch(void* const* d_in, const int* in_sizes, int n_in,
                              void* d_out, int out_size, void* d_ws, size_t ws_size,
                              hipStream_t stream) {
    const float* x  = (const float*)d_in[0];
    const float* fb = (const float*)d_in[1];
    float*       y  = (float*)d_out;
    (void)in_sizes; (void)n_in; (void)out_size; (void)d_ws; (void)ws_size;

    // 126 phase-tiles per row, 8 waves per 256-thread block -> 16 blocks in x.
    dim3 grid((PTILES + 7) / 8, NB, 1);
    dim3 block(256, 1, 1);
    allpass_wmma_kernel<<<grid, block, 0, stream>>>(x, fb, y);
}